// GraphAttention_52140902974095
// MI455X (gfx1250) — compile-verified
//
#include <hip/hip_runtime.h>

// ---------------------------------------------------------------------------
// GraphAttention on gfx1250 (MI455X): gather -> score -> segment softmax ->
// attention-weighted x@W via V_WMMA_F32_16X16X4_F32 -> scatter-add -> relu.
// ---------------------------------------------------------------------------

typedef float v2f __attribute__((ext_vector_type(2)));
typedef float v8f __attribute__((ext_vector_type(8)));

#define DFEAT 128
#define KPAD 260                 // permuted x row: [src 0..127][dst 128..255][dist 256..257][0,0]
#define KK_STEPS 65              // 260 / 4
#define WPK_FLOATS (KK_STEPS * 512)   // packed w_embed in LDS: [kk][n][g][h]
#define XT_FLOATS (16 * KPAD)    // per-wave A tile (16 edges x 260)

// Monotonic float<->uint encoding so atomicMax on u32 == float max.
__device__ __forceinline__ unsigned enc_f32(float f) {
    unsigned u = __float_as_uint(f);
    return (u & 0x80000000u) ? ~u : (u | 0x80000000u);
}
__device__ __forceinline__ float dec_f32(unsigned u) {
    return (u & 0x80000000u) ? __uint_as_float(u ^ 0x80000000u) : __uint_as_float(~u);
}

// ---------------------------------------------------------------------------
__global__ void init_kernel(float* __restrict__ out, int out_n,
                            unsigned* __restrict__ segmax, float* __restrict__ denom,
                            int ndst) {
    int i = blockIdx.x * blockDim.x + threadIdx.x;
    int stride = gridDim.x * blockDim.x;
    for (int t = i; t < out_n; t += stride) out[t] = 0.0f;
    for (int t = i; t < ndst; t += stride) {
        segmax[t] = 0x007FFFFFu;   // enc(-inf)
        denom[t] = 0.0f;
    }
}

// ---------------------------------------------------------------------------
// 8 lanes per edge: coalesced float4 reads of the two feature rows,
// shfl-xor reduction, LeakyReLU(0.2), atomic segment-max.
__global__ void score_kernel(const float* __restrict__ srcf, const int* __restrict__ esrc,
                             const float* __restrict__ dstf, const int* __restrict__ edst,
                             const float* __restrict__ dist, const float* __restrict__ w_att,
                             float* __restrict__ score, unsigned* __restrict__ segmax, int E) {
    int gid = blockIdx.x * blockDim.x + threadIdx.x;
    int edge = gid >> 3;
    int q = gid & 7;
    if (edge >= E) return;
    int is = esrc[edge];
    int id = edst[edge];
    const float4* s4 = (const float4*)(srcf + (size_t)is * DFEAT);
    const float4* d4 = (const float4*)(dstf + (size_t)id * DFEAT);
    const float4* wa4 = (const float4*)w_att;
    float acc = 0.0f;
#pragma unroll
    for (int t = 0; t < 4; ++t) {
        int f = q * 4 + t;
        float4 v = s4[f];
        float4 w = wa4[f];
        acc += v.x * w.x + v.y * w.y + v.z * w.z + v.w * w.w;
        float4 v2 = d4[f];
        const float* wd = w_att + 130 + f * 4;   // dst block of w_att
        acc += v2.x * wd[0] + v2.y * wd[1] + v2.z * wd[2] + v2.w * wd[3];
    }
    if (q == 0)
        acc += dist[(size_t)edge * 2] * w_att[128] + dist[(size_t)edge * 2 + 1] * w_att[129];
    acc += __shfl_xor(acc, 1);
    acc += __shfl_xor(acc, 2);
    acc += __shfl_xor(acc, 4);
    if (q == 0) {
        float sc = (acc >= 0.0f) ? acc : 0.2f * acc;
        score[edge] = sc;
        atomicMax(segmax + id, enc_f32(sc));
    }
}

// ---------------------------------------------------------------------------
__global__ void exp_kernel(float* __restrict__ score, const int* __restrict__ edst,
                           const unsigned* __restrict__ segmax, float* __restrict__ denom,
                           int E) {
    int i = blockIdx.x * blockDim.x + threadIdx.x;
    if (i >= E) return;
    int id = edst[i];
    float mx = dec_f32(segmax[id]);
    float e = __expf(score[i] - mx);
    score[i] = e;                 // overwrite score slot with e
    atomicAdd(denom + id, e);
}

// ---------------------------------------------------------------------------
// Main kernel: per wave, 16 edges x 128 outputs via V_WMMA_F32_16X16X4_F32.
// Attention is folded into A:  out[dst] += (attn*x) @ W.
// LDS: packed W (bank-conflict-free B layout) + per-wave A tiles.
__global__ void __launch_bounds__(256)
gat_wmma_kernel(const float* __restrict__ srcf, const int* __restrict__ esrc,
                const float* __restrict__ dstf, const int* __restrict__ edst,
                const float* __restrict__ dist, const float* __restrict__ w_embed,
                const float* __restrict__ eval, const float* __restrict__ denom,
                float* __restrict__ out, int E, int ntiles) {
    extern __shared__ float smem[];
    const int tid = threadIdx.x;
    const int lane = tid & 31;
    const int wave = tid >> 5;
    const int g = lane >> 4;     // half-wave group (0/1)
    const int m = lane & 15;

    // ---- Stage w_embed into LDS, pre-swizzled for the B operand layout.
    // idx = kk*512 + n*4 + g*2 + h  holds  W[col = 4*kk + 2*g + h][n],
    // with x-columns permuted: col<128 -> src (w row col), 128..255 -> dst
    // (w row col+2), 256/257 -> dist (w rows 128/129), >=258 -> zero pad.
    for (int idx = tid; idx < WPK_FLOATS; idx += 256) {
        int kk = idx >> 9;
        int r = idx & 511;
        int n = r >> 2;
        int q = r & 3;
        int c = 4 * kk + q;
        float v = 0.0f;
        if (c < 128)       v = w_embed[(size_t)c * DFEAT + n];
        else if (c < 256)  v = w_embed[(size_t)(c + 2) * DFEAT + n];
        else if (c == 256) v = w_embed[(size_t)128 * DFEAT + n];
        else if (c == 257) v = w_embed[(size_t)129 * DFEAT + n];
        smem[idx] = v;
    }
    __syncthreads();

    float* xt = smem + WPK_FLOATS + wave * XT_FLOATS;
    const int gw = blockIdx.x * 8 + wave;
    const int nw = gridDim.x * 8;

    for (int tile = gw; tile < ntiles; tile += nw) {
        const int tb = tile * 16;

        // ---- Stage A tile: 16 edges, rows scaled by attn, columns permuted.
        for (int e = 0; e < 16; ++e) {
            int edge = tb + e;
            float attn = 0.0f;
            int is = 0, id = 0;
            if (edge < E) {
                is = esrc[edge];
                id = edst[edge];
                attn = eval[edge] / denom[id];
            }
            float4 s = ((const float4*)(srcf + (size_t)is * DFEAT))[lane];
            float4 d = ((const float4*)(dstf + (size_t)id * DFEAT))[lane];
            float4* row = (float4*)(xt + e * KPAD);
            float4 sv; sv.x = s.x * attn; sv.y = s.y * attn; sv.z = s.z * attn; sv.w = s.w * attn;
            float4 dv; dv.x = d.x * attn; dv.y = d.y * attn; dv.z = d.z * attn; dv.w = d.w * attn;
            row[lane] = sv;          // cols 0..127   (16B aligned: e*1040 bytes)
            row[32 + lane] = dv;     // cols 128..255 (+512B, still 16B aligned)
            if (lane == 0) {
                float d0 = (edge < E) ? dist[(size_t)edge * 2] : 0.0f;
                float d1 = (edge < E) ? dist[(size_t)edge * 2 + 1] : 0.0f;
                xt[e * KPAD + 256] = d0 * attn;
                xt[e * KPAD + 257] = d1 * attn;
                xt[e * KPAD + 258] = 0.0f;
                xt[e * KPAD + 259] = 0.0f;
            }
        }
        // Per-wave LDS ops are in-order, but be explicit before WMMA consumes.
        asm volatile("s_wait_dscnt 0" ::: "memory");

        // ---- K loop: 65 steps of V_WMMA_F32_16X16X4_F32 across 8 N-tiles.
        v8f acc[8] = {};
        for (int kk = 0; kk < KK_STEPS; ++kk) {
            // A 16x4: lane m = row M, group g picks K pair (k0+2g, k0+2g+1).
            v2f a = *(const v2f*)(xt + m * KPAD + kk * 4 + g * 2);
            const float* wb = smem + kk * 512 + g * 2;
#pragma unroll
            for (int nt = 0; nt < 8; ++nt) {
                // B 4x16: lane m = col N (within tile), group g picks K pair.
                v2f b = *(const v2f*)(wb + (nt * 16 + m) * 4);
                acc[nt] = __builtin_amdgcn_wmma_f32_16x16x4_f32(
                    false, a, false, b, (short)0, acc[nt], false, false);
            }
        }

        // ---- Scatter: C layout lane L reg r -> M = 8*(L/16)+r, N = 16*nt + (L%16).
#pragma unroll
        for (int r = 0; r < 8; ++r) {
            int M = g * 8 + r;
            int edge = tb + M;
            if (edge < E) {
                float* orow = out + (size_t)edst[edge] * DFEAT;
#pragma unroll
                for (int nt = 0; nt < 8; ++nt)
                    atomicAdd(orow + nt * 16 + m, acc[nt][r]);
            }
        }
    }
}

// ---------------------------------------------------------------------------
__global__ void relu_kernel(float* __restrict__ out, int n) {
    int i = blockIdx.x * blockDim.x + threadIdx.x;
    int stride = gridDim.x * blockDim.x;
    for (int t = i; t < n; t += stride) out[t] = fmaxf(out[t], 0.0f);
}

// ---------------------------------------------------------------------------
extern "C" void kernel_launch(void* const* d_in, const int* in_sizes, int n_in,
                              void* d_out, int out_size, void* d_ws, size_t ws_size,
                              hipStream_t stream) {
    const float* srcf    = (const float*)d_in[0];
    const int*   esrc    = (const int*)d_in[1];
    const float* dstf    = (const float*)d_in[2];
    const int*   edst    = (const int*)d_in[3];
    const float* dist    = (const float*)d_in[4];
    const float* w_att   = (const float*)d_in[5];
    const float* w_embed = (const float*)d_in[6];
    float* out = (float*)d_out;

    const int E = in_sizes[1];          // edge_src_idx count
    const int ndst = out_size / DFEAT;  // 50000

    // Workspace: [E] score->e | [ndst] segmax(u32) | [ndst] denom(f32)
    float* score = (float*)d_ws;
    unsigned* segmax = (unsigned*)(score + E);
    float* denom = (float*)(segmax + ndst);

    init_kernel<<<256, 256, 0, stream>>>(out, out_size, segmax, denom, ndst);

    int sblocks = (E * 8 + 255) / 256;
    score_kernel<<<sblocks, 256, 0, stream>>>(srcf, esrc, dstf, edst, dist, w_att,
                                              score, segmax, E);

    exp_kernel<<<(E + 255) / 256, 256, 0, stream>>>(score, edst, segmax, denom, E);

    int ntiles = (E + 15) / 16;
    int wblocks = (ntiles + 7) / 8;
    if (wblocks > 512) wblocks = 512;
    if (wblocks < 1) wblocks = 1;
    size_t shbytes = (size_t)(WPK_FLOATS + 8 * XT_FLOATS) * sizeof(float);
    gat_wmma_kernel<<<wblocks, 256, shbytes, stream>>>(srcf, esrc, dstf, edst, dist,
                                                       w_embed, score, denom, out,
                                                       E, ntiles);

    relu_kernel<<<(out_size + 255) / 256, 256, 0, stream>>>(out, out_size);
}